// GraphAutoencoder_6760278524061
// MI455X (gfx1250) — compile-verified
//
#include <hip/hip_runtime.h>
#include <hip/hip_bf16.h>

#define NODES 16384
#define EDGES 524288
#define INDIM 256

typedef float v2f __attribute__((ext_vector_type(2)));
typedef float v8f __attribute__((ext_vector_type(8)));

#if defined(__gfx1250__)
#define WMMA_F32(a, b, c) \
    __builtin_amdgcn_wmma_f32_16x16x4_f32(false, (a), false, (b), (short)0, (c), false, false)
#else
static __device__ inline v8f wmma_host_stub(v2f a, v2f b, v8f c) { c[0] += a.x * b.x; return c; }
#define WMMA_F32(a, b, c) wmma_host_stub((a), (b), (c))
#endif

// ---------------------------------------------------------------- utilities
__global__ void fill_kernel(float* __restrict__ p, float v, int n) {
    int i = blockIdx.x * blockDim.x + threadIdx.x;
    if (i < n) p[i] = v;
}

// deg[dst] += 1 per edge (deg pre-filled with 1.0 for the self loop)
__global__ void deg_kernel(const long long* __restrict__ dst, float* __restrict__ deg, int e) {
    int i = blockIdx.x * blockDim.x + threadIdx.x;
    if (i < e) atomicAdd(&deg[(int)dst[i]], 1.0f);
}

// deg -> deg^{-1/2} in place
__global__ void dinv_kernel(float* __restrict__ d, int n) {
    int i = blockIdx.x * blockDim.x + threadIdx.x;
    if (i < n) d[i] = rsqrtf(d[i]);
}

// ---------------------------------------------------------------- WMMA GEMMs
// C[M,N] = A[M,K] @ B[K,N], fp32 row-major, one 16x16 tile per wave,
// K stepped by 4 via V_WMMA_F32_16X16X4_F32. M,N,K multiples of 16/4.
__global__ void gemm_nn_wmma(const float* __restrict__ A, const float* __restrict__ B,
                             float* __restrict__ C, int M, int K, int N) {
    int wave   = blockIdx.x * (blockDim.x >> 5) + (threadIdx.x >> 5);
    int tilesN = N >> 4;
    int tm = wave / tilesN;
    int tn = wave - tm * tilesN;
    if (tm >= (M >> 4)) return;                // wave-uniform: EXEC stays all-ones

    int lane = threadIdx.x & 31;
    int mr   = lane & 15;                      // M (for A) / N (for B) within tile
    int kb   = (lane >> 4) << 1;               // K sub-offset: 0 (lanes 0-15) or 2

    const float* Arow = A + (size_t)(tm * 16 + mr) * K;   // A[M=mr row]
    const float* Bcol = B + tn * 16 + mr;                 // B[:, N=mr col]

    v8f acc = {};
    for (int k = 0; k < K; k += 4) {
        v2f a = *(const v2f*)(Arow + k + kb);              // 8B aligned -> b64 load
        v2f b;
        b.x = Bcol[(size_t)(k + kb) * N];
        b.y = Bcol[(size_t)(k + kb + 1) * N];
        acc = WMMA_F32(a, b, acc);
    }

    int rowbase = tm * 16 + ((lane >> 4) << 3);   // M = r + 8*(lane>=16)
    int col     = tn * 16 + mr;
#pragma unroll
    for (int r = 0; r < 8; ++r)
        C[(size_t)(rowbase + r) * N + col] = acc[r];
}

// C[Nn,Nn] = S @ S^T, S row-major [Nn,K].
// Register-blocked 2x4 tiles (32x64 output strip) per wave: cuts L2 reads 2.7x
// vs 1 tile/wave and gives 8 independent accumulators so same-acc WMMA RAW
// dependencies are 8 instructions apart (hides XDL latency).
// Output (~1GB) streamed with non-temporal stores to keep it out of L2.
__global__ void adj_nt_wmma(const float* __restrict__ S, float* __restrict__ C,
                            int Nn, int K) {
    int wave    = blockIdx.x * (blockDim.x >> 5) + (threadIdx.x >> 5);
    int stripsN = Nn >> 6;                         // 64-wide col strips
    int sm = wave / stripsN;                       // 32-tall row strip
    int sn = wave - sm * stripsN;
    if (sm >= (Nn >> 5)) return;                   // wave-uniform exit

    int lane = threadIdx.x & 31;
    int mr   = lane & 15;
    int kb   = (lane >> 4) << 1;

    const float* Ar0 = S + (size_t)(sm * 32 + mr) * K;
    const float* Ar1 = Ar0 + (size_t)16 * K;
    const float* Br0 = S + (size_t)(sn * 64 + mr) * K;
    const float* Br1 = Br0 + (size_t)16 * K;
    const float* Br2 = Br0 + (size_t)32 * K;
    const float* Br3 = Br0 + (size_t)48 * K;

    v8f acc[8] = {};                               // acc[i*4+j]: row tile i, col tile j
    for (int k = 0; k < K; k += 4) {
        int o = k + kb;                            // 8B aligned offsets -> b64 loads
        v2f a0 = *(const v2f*)(Ar0 + o);
        v2f a1 = *(const v2f*)(Ar1 + o);
        v2f b0 = *(const v2f*)(Br0 + o);
        v2f b1 = *(const v2f*)(Br1 + o);
        v2f b2 = *(const v2f*)(Br2 + o);
        v2f b3 = *(const v2f*)(Br3 + o);
        acc[0] = WMMA_F32(a0, b0, acc[0]);
        acc[1] = WMMA_F32(a0, b1, acc[1]);
        acc[2] = WMMA_F32(a0, b2, acc[2]);
        acc[3] = WMMA_F32(a0, b3, acc[3]);
        acc[4] = WMMA_F32(a1, b0, acc[4]);
        acc[5] = WMMA_F32(a1, b1, acc[5]);
        acc[6] = WMMA_F32(a1, b2, acc[6]);
        acc[7] = WMMA_F32(a1, b3, acc[7]);
    }

#pragma unroll
    for (int i = 0; i < 2; ++i) {
        int rowbase = sm * 32 + i * 16 + ((lane >> 4) << 3);
#pragma unroll
        for (int j = 0; j < 4; ++j) {
            int col = sn * 64 + j * 16 + mr;
#pragma unroll
            for (int r = 0; r < 8; ++r)
                __builtin_nontemporal_store(acc[i * 4 + j][r],
                                            &C[(size_t)(rowbase + r) * Nn + col]);
        }
    }
}

// ---------------------------------------------------------------- aggregation
// agg[dst] += h[src] * dinv[src]*dinv[dst], 4 features per thread
__global__ void scatter_kernel(const float* __restrict__ h,
                               const long long* __restrict__ src,
                               const long long* __restrict__ dst,
                               const float* __restrict__ dinv,
                               float* __restrict__ agg, int outdim) {
    long long tid    = (long long)blockIdx.x * blockDim.x + threadIdx.x;
    int       chunks = outdim >> 2;
    long long total  = (long long)EDGES * chunks;
    if (tid >= total) return;
    int e = (int)(tid / chunks);
    int f = (int)(tid - (long long)e * chunks) << 2;
    int s = (int)src[e];
    int d = (int)dst[e];
    float en = dinv[s] * dinv[d];
    const float4 v = *(const float4*)(h + (size_t)s * outdim + f);
    float* ap = agg + (size_t)d * outdim + f;
    atomicAdd(ap + 0, v.x * en);
    atomicAdd(ap + 1, v.y * en);
    atomicAdd(ap + 2, v.z * en);
    atomicAdd(ap + 3, v.w * en);
}

// out = agg + h*dinv^2 + bias  (+ optional relu)
__global__ void combine_kernel(const float* __restrict__ agg, const float* __restrict__ h,
                               const float* __restrict__ dinv, const float* __restrict__ bias,
                               float* __restrict__ out, int outdim, int relu) {
    int i = blockIdx.x * blockDim.x + threadIdx.x;
    if (i >= NODES * outdim) return;
    int n = i / outdim;
    int f = i - n * outdim;
    float di = dinv[n];
    float v  = agg[i] + h[i] * di * di + bias[f];
    out[i] = relu ? fmaxf(v, 0.0f) : v;
}

// ---------------------------------------------------------------- host side
static void run_gcn(const float* xin, const float* W, const float* bias,
                    int indim, int outdim, int relu,
                    const long long* src, const long long* dst, const float* dinv,
                    float* h, float* agg, float* out, hipStream_t stream) {
    int tiles = (NODES / 16) * (outdim / 16);
    gemm_nn_wmma<<<(tiles + 7) / 8, 256, 0, stream>>>(xin, W, h, NODES, indim, outdim);

    int nelem = NODES * outdim;
    fill_kernel<<<(nelem + 255) / 256, 256, 0, stream>>>(agg, 0.0f, nelem);

    long long total = (long long)EDGES * (outdim >> 2);
    scatter_kernel<<<(unsigned)((total + 255) / 256), 256, 0, stream>>>(h, src, dst, dinv,
                                                                        agg, outdim);
    combine_kernel<<<(nelem + 255) / 256, 256, 0, stream>>>(agg, h, dinv, bias, out,
                                                            outdim, relu);
}

extern "C" void kernel_launch(void* const* d_in, const int* in_sizes, int n_in,
                              void* d_out, int out_size, void* d_ws, size_t ws_size,
                              hipStream_t stream) {
    (void)in_sizes; (void)n_in; (void)out_size; (void)ws_size;

    const float*     x    = (const float*)d_in[0];
    const long long* ei   = (const long long*)d_in[1];   // int64 [2, E]
    const long long* src  = ei;
    const long long* dst  = ei + EDGES;
    const float* W_g1 = (const float*)d_in[2];  const float* b_g1 = (const float*)d_in[3];
    const float* W_g2 = (const float*)d_in[4];  const float* b_g2 = (const float*)d_in[5];
    const float* W_a1 = (const float*)d_in[6];  const float* b_a1 = (const float*)d_in[7];
    const float* W_a2 = (const float*)d_in[8];  const float* b_a2 = (const float*)d_in[9];
    const float* W_s  = (const float*)d_in[10]; const float* b_s  = (const float*)d_in[11];

    float* xrec = (float*)d_out;                          // [N, 256]
    float* adj  = (float*)d_out + (size_t)NODES * INDIM;  // [N, N]

    float* p    = (float*)d_ws;
    float* dinv = p; p += NODES;                   // deg, then deg^{-1/2} in place
    float* h    = p; p += (size_t)NODES * 256;     // GEMM output scratch
    float* agg  = p; p += (size_t)NODES * 256;     // segment-sum scratch
    float* z1   = p; p += (size_t)NODES * 128;     // relu(gcn g1)
    float* z2   = p; p += (size_t)NODES * 64;      // relu(gcn g2) = z
    float* a1   = p; p += (size_t)NODES * 128;     // relu(gcn a1)
    float* sb   = p; p += (size_t)NODES * 64;      // relu(gcn s)

    // degrees -> dinv
    fill_kernel<<<(NODES + 255) / 256, 256, 0, stream>>>(dinv, 1.0f, NODES);
    deg_kernel<<<(EDGES + 255) / 256, 256, 0, stream>>>(dst, dinv, EDGES);
    dinv_kernel<<<(NODES + 255) / 256, 256, 0, stream>>>(dinv, NODES);

    // encoder
    run_gcn(x,  W_g1, b_g1, 256, 128, 1, src, dst, dinv, h, agg, z1,   stream);
    run_gcn(z1, W_g2, b_g2, 128,  64, 1, src, dst, dinv, h, agg, z2,   stream);
    // attribute decoder
    run_gcn(z2, W_a1, b_a1,  64, 128, 1, src, dst, dinv, h, agg, a1,   stream);
    run_gcn(a1, W_a2, b_a2, 128, 256, 0, src, dst, dinv, h, agg, xrec, stream);
    // structure decoder
    run_gcn(z2, W_s,  b_s,   64,  64, 1, src, dst, dinv, h, agg, sb,   stream);

    // adj_rec = sb @ sb^T  (512 x 256 strips of 32x64, 8 waves per block)
    long long strips = (long long)(NODES / 32) * (NODES / 64);
    adj_nt_wmma<<<(unsigned)(strips / 8), 256, 0, stream>>>(sb, adj, NODES, 64);
}